// RenyiDivergenceLoss_34359738368289
// MI455X (gfx1250) — compile-verified
//
#include <hip/hip_runtime.h>
#include <hip/hip_bf16.h>
#include <math.h>

typedef float v2f __attribute__((ext_vector_type(2)));
typedef float v8f __attribute__((ext_vector_type(8)));

#define NROWS   8192
#define DIM     32
#define TILES   (NROWS / 16)   // 512
#define NWAVES  4
#define KNN_K   3

// ---------------------------------------------------------------------------
// Row squared-norms: one thread per row, 8x float4 loads.
// ---------------------------------------------------------------------------
__global__ void norms_kernel(const float* __restrict__ A, float* __restrict__ out, int n)
{
    int i = blockIdx.x * blockDim.x + threadIdx.x;
    if (i >= n) return;
    const float4* r = (const float4*)(A + i * DIM);
    float s = 0.0f;
#pragma unroll
    for (int j = 0; j < DIM / 4; ++j) {
        float4 v = r[j];
        s += v.x * v.x + v.y * v.y + v.z * v.z + v.w * v.w;
    }
    out[i] = s;
}

// Branchless sorted-4 insert (ascending s0<=s1<=s2<=s3), keeps 4 smallest.
__device__ __forceinline__ void insert4(float v, float& s0, float& s1, float& s2, float& s3)
{
    float n3 = fminf(s3, fmaxf(s2, v));
    float n2 = fminf(s2, fmaxf(s1, v));
    float n1 = fminf(s1, fmaxf(s0, v));
    float n0 = fminf(s0, v);
    s0 = n0; s1 = n1; s2 = n2; s3 = n3;
}

// Extract even/odd float pairs of a float4 as WMMA v2f operands.
__device__ __forceinline__ v2f pair_lo(float4 v) { v2f r; r.x = v.x; r.y = v.y; return r; }
__device__ __forceinline__ v2f pair_hi(float4 v) { v2f r; r.x = v.z; r.y = v.w; return r; }

// Load one lane's contiguous 16-float K-slice (half-row) as 4 float4s.
// 32-bit element offset keeps address math in one VGPR (saddr-form loads).
// K-permutation: lane-half h owns K in [16h, 16h+16); valid because WMMA sums
// over K order-independently and A/B use the identical permutation.
__device__ __forceinline__ void load_half_row(const float* __restrict__ base, int foff, float4* dst)
{
    const float4* p = (const float4*)(base + foff);
    dst[0] = p[0]; dst[1] = p[1]; dst[2] = p[2]; dst[3] = p[3];
}

// 8-step K=32 WMMA chain: c += A-slice x B-slice.
__device__ __forceinline__ v8f wmma_k32(const v2f* __restrict__ ap, const float4* __restrict__ bf)
{
    v8f c = {};
    c = __builtin_amdgcn_wmma_f32_16x16x4_f32(false, ap[0], false, pair_lo(bf[0]), (short)0, c, false, false);
    c = __builtin_amdgcn_wmma_f32_16x16x4_f32(false, ap[1], false, pair_hi(bf[0]), (short)0, c, false, false);
    c = __builtin_amdgcn_wmma_f32_16x16x4_f32(false, ap[2], false, pair_lo(bf[1]), (short)0, c, false, false);
    c = __builtin_amdgcn_wmma_f32_16x16x4_f32(false, ap[3], false, pair_hi(bf[1]), (short)0, c, false, false);
    c = __builtin_amdgcn_wmma_f32_16x16x4_f32(false, ap[4], false, pair_lo(bf[2]), (short)0, c, false, false);
    c = __builtin_amdgcn_wmma_f32_16x16x4_f32(false, ap[5], false, pair_hi(bf[2]), (short)0, c, false, false);
    c = __builtin_amdgcn_wmma_f32_16x16x4_f32(false, ap[6], false, pair_lo(bf[3]), (short)0, c, false, false);
    c = __builtin_amdgcn_wmma_f32_16x16x4_f32(false, ap[7], false, pair_hi(bf[3]), (short)0, c, false, false);
    return c;
}

// ---------------------------------------------------------------------------
// knn_nu: for each row j of Y, nu[j] = 4th-smallest distance to all rows of Y
// (self included, matching the reference's knn[:, k] with k=3).
// One workgroup = one 16-row tile; 4 waves split the 512 column tiles.
// B operands double-buffered so the insertion VALU phase hides load latency.
// Wave index is readfirstlane'd so the tile loop is scalar (SALU control,
// EXEC provably all-ones around the WMMAs).
// ---------------------------------------------------------------------------
__global__ __launch_bounds__(32 * NWAVES) void knn_nu_kernel(
    const float* __restrict__ Y, const float* __restrict__ nY,
    float* __restrict__ nu_out, float* __restrict__ rhosq_out)
{
    const int rowBase = blockIdx.x * 16;
    const int lane = threadIdx.x & 31;
    const int wave = __builtin_amdgcn_readfirstlane(threadIdx.x >> 5);  // SGPR, wave-uniform
    const int lr = lane & 15;      // M for A-operand, N for B/C/D
    const int hk = lane >> 4;      // K-half selector: this lane owns K [16hk,16hk+16)
    const int laneoff = lr * DIM + 16 * hk;   // per-lane element offset within a tile

    // A operand (fixed for this block): contiguous half-row, 4 x b128.
    float4 af[4];
    load_half_row(Y, rowBase * DIM + laneoff, af);
    v2f ap[8];
#pragma unroll
    for (int cidx = 0; cidx < 4; ++cidx) {
        ap[2 * cidx]     = pair_lo(af[cidx]);
        ap[2 * cidx + 1] = pair_hi(af[cidx]);
    }

    // Squared-norms for the 8 accumulator rows this lane covers: M = r + 8*hk
    const float4* nrp = (const float4*)(nY + rowBase + 8 * hk);
    float4 nr0 = nrp[0], nr1 = nrp[1];
    float nrow[8] = { nr0.x, nr0.y, nr0.z, nr0.w, nr1.x, nr1.y, nr1.z, nr1.w };

    float s0[8], s1[8], s2[8], s3[8];
#pragma unroll
    for (int r = 0; r < 8; ++r) { s0[r] = 3.4e38f; s1[r] = 3.4e38f; s2[r] = 3.4e38f; s3[r] = 3.4e38f; }

    // Prime the pipeline with the first column tile.
    float4 bcur[4];
    load_half_row(Y, wave * 16 * DIM + laneoff, bcur);
    float ncur = nY[wave * 16 + lr];

    for (int ct = wave; ct < TILES; ct += NWAVES) {
        // Prefetch next tile (clamped; scalar index).
        const int nt = (ct + NWAVES < TILES) ? (ct + NWAVES) : wave;
        float4 bnext[4];
        load_half_row(Y, nt * 16 * DIM + laneoff, bnext);
        float nnext = nY[nt * 16 + lr];

        v8f c = wmma_k32(ap, bcur);

#pragma unroll
        for (int r = 0; r < 8; ++r) {
            float v = nrow[r] + ncur - 2.0f * c[r];   // squared distance (unclamped)
            insert4(v, s0[r], s1[r], s2[r], s3[r]);
        }

        bcur[0] = bnext[0]; bcur[1] = bnext[1]; bcur[2] = bnext[2]; bcur[3] = bnext[3];
        ncur = nnext;
    }

    // Merge per-lane candidate lists across lanes & waves via LDS.
    __shared__ float smem[16][NWAVES * 16][4];
#pragma unroll
    for (int r = 0; r < 8; ++r) {
        const int row = r + 8 * hk;
        float* p = smem[row][wave * 16 + lr];
        p[0] = s0[r]; p[1] = s1[r]; p[2] = s2[r]; p[3] = s3[r];
    }
    __syncthreads();

    if (threadIdx.x < 16) {
        const int row = threadIdx.x;
        float t0 = 3.4e38f, t1 = 3.4e38f, t2 = 3.4e38f, t3 = 3.4e38f;
        for (int s = 0; s < NWAVES * 16; ++s) {
            const float* p = smem[row][s];
            insert4(p[0], t0, t1, t2, t3);
            insert4(p[1], t0, t1, t2, t3);
            insert4(p[2], t0, t1, t2, t3);
            insert4(p[3], t0, t1, t2, t3);
        }
        // 4th smallest (index k=3) of the full row; clamp as sqrt(max(sq,1e-12)).
        // Order statistics commute with the monotone clamp, so clamping only
        // the selected value is exact.
        float nusq = fmaxf(t3, 1e-12f);
        float nu = sqrtf(nusq);
        nu_out[rowBase + row] = nu;
        rhosq_out[rowBase + row] = nusq;   // rho == nu since nu >= 1e-6 >> 1e-20
    }
}

// ---------------------------------------------------------------------------
// count: k_p[j] = #{ i : dist(X_i, Y_j) <= rho[j] }.
// One workgroup = one 16-column tile of Y (B operand hoisted); 4 waves split
// the 512 row tiles of X. Each lane owns one column -> lane-local counting.
// A operands + X norms double-buffered; scalar loop control.
// ---------------------------------------------------------------------------
__global__ __launch_bounds__(32 * NWAVES) void count_kernel(
    const float* __restrict__ X, const float* __restrict__ nX,
    const float* __restrict__ Y, const float* __restrict__ nYv,
    const float* __restrict__ rhosq, float* __restrict__ kp)
{
    const int colBase = blockIdx.x * 16;
    const int lane = threadIdx.x & 31;
    const int wave = __builtin_amdgcn_readfirstlane(threadIdx.x >> 5);  // SGPR
    const int lr = lane & 15;
    const int hk = lane >> 4;
    const int laneoff = lr * DIM + 16 * hk;

    // B operand (fixed): column n = lr of Y, contiguous K half-row.
    float4 bf[4];
    load_half_row(Y, colBase * DIM + laneoff, bf);

    const float ncol = nYv[colBase + lr];
    const float rho2 = rhosq[colBase + lr];

    // Prime pipeline with first row tile.
    float4 acur[4];
    load_half_row(X, wave * 16 * DIM + laneoff, acur);
    const float4* np0 = (const float4*)(nX + wave * 16 + 8 * hk);
    float4 nc0 = np0[0], nc1 = np0[1];

    int cnt = 0;
    for (int rt = wave; rt < TILES; rt += NWAVES) {
        const int nt = (rt + NWAVES < TILES) ? (rt + NWAVES) : wave;
        float4 anext[4];
        load_half_row(X, nt * 16 * DIM + laneoff, anext);
        const float4* npn = (const float4*)(nX + nt * 16 + 8 * hk);
        float4 nn0 = npn[0], nn1 = npn[1];

        v2f ap[8];
#pragma unroll
        for (int cidx = 0; cidx < 4; ++cidx) {
            ap[2 * cidx]     = pair_lo(acur[cidx]);
            ap[2 * cidx + 1] = pair_hi(acur[cidx]);
        }
        v8f c = wmma_k32(ap, bf);

        float nrow[8] = { nc0.x, nc0.y, nc0.z, nc0.w, nc1.x, nc1.y, nc1.z, nc1.w };
#pragma unroll
        for (int r = 0; r < 8; ++r) {
            float sq = nrow[r] + ncol - 2.0f * c[r];
            sq = fmaxf(sq, 1e-12f);
            cnt += (sq <= rho2) ? 1 : 0;
        }

        acur[0] = anext[0]; acur[1] = anext[1]; acur[2] = anext[2]; acur[3] = anext[3];
        nc0 = nn0; nc1 = nn1;
    }

    __shared__ int scnt[NWAVES][32];
    scnt[wave][lane] = cnt;
    __syncthreads();
    if (threadIdx.x < 16) {
        int tot = 0;
#pragma unroll
        for (int w = 0; w < NWAVES; ++w)
            tot += scnt[w][threadIdx.x] + scnt[w][threadIdx.x + 16];
        kp[colBase + threadIdx.x] = (float)tot;
    }
}

// ---------------------------------------------------------------------------
// finalize: scalar reduction for both directions; out = max(0, log r_p, log r_q)
// ---------------------------------------------------------------------------
__global__ __launch_bounds__(256) void finalize_kernel(
    const float* __restrict__ kp1, const float* __restrict__ nuQ,
    const float* __restrict__ kp2, const float* __restrict__ nuP,
    float* __restrict__ out)
{
    const int m = NROWS;
    const float eps = 1e-20f;
    const float kq_sum = 3.0f * (float)m + eps;
    __shared__ double red[256];
    const int tid = threadIdx.x;

    // kp sums
    double a1 = 0.0, a2 = 0.0;
    for (int j = tid; j < m; j += 256) { a1 += (double)kp1[j]; a2 += (double)kp2[j]; }
    red[tid] = a1; __syncthreads();
    for (int off = 128; off > 0; off >>= 1) { if (tid < off) red[tid] += red[tid + off]; __syncthreads(); }
    const float kp_sum1 = (float)red[0] + eps;
    __syncthreads();
    red[tid] = a2; __syncthreads();
    for (int off = 128; off > 0; off >>= 1) { if (tid < off) red[tid] += red[tid + off]; __syncthreads(); }
    const float kp_sum2 = (float)red[0] + eps;
    __syncthreads();

    double r1 = 0.0, r2 = 0.0;
    const float qscale = 3.0f / kq_sum;
    for (int j = tid; j < m; j += 256) {
        {   // direction 1: estimate(P, Q)
            float nu = nuQ[j];
            float n2 = nu * nu, n4 = n2 * n2, n8 = n4 * n4, n16 = n8 * n8;
            float n32 = n16 * n16;                       // rho^d, d=32
            float inv = 1.0f / (n32 + eps);
            float p = fminf(fmaxf(kp1[j] / kp_sum1 * inv, eps), 1e10f);
            float q = fminf(fmaxf(qscale * inv, eps), 1e10f);
            r1 += (double)(p / q) * (double)qscale;
        }
        {   // direction 2: estimate(Q, P)
            float nu = nuP[j];
            float n2 = nu * nu, n4 = n2 * n2, n8 = n4 * n4, n16 = n8 * n8;
            float n32 = n16 * n16;
            float inv = 1.0f / (n32 + eps);
            float p = fminf(fmaxf(kp2[j] / kp_sum2 * inv, eps), 1e10f);
            float q = fminf(fmaxf(qscale * inv, eps), 1e10f);
            r2 += (double)(p / q) * (double)qscale;
        }
    }
    red[tid] = r1; __syncthreads();
    for (int off = 128; off > 0; off >>= 1) { if (tid < off) red[tid] += red[tid + off]; __syncthreads(); }
    const double R1 = red[0];
    __syncthreads();
    red[tid] = r2; __syncthreads();
    for (int off = 128; off > 0; off >>= 1) { if (tid < off) red[tid] += red[tid + off]; __syncthreads(); }
    const double R2 = red[0];

    if (tid == 0) {
        float v1 = logf((float)R1);   // alpha-1 == 1
        float v2 = logf((float)R2);
        out[0] = fmaxf(0.0f, fmaxf(v1, v2));
    }
}

// ---------------------------------------------------------------------------
extern "C" void kernel_launch(void* const* d_in, const int* in_sizes, int n_in,
                              void* d_out, int out_size, void* d_ws, size_t ws_size,
                              hipStream_t stream)
{
    const float* P = (const float*)d_in[0];
    const float* Q = (const float*)d_in[1];
    float* out = (float*)d_out;

    float* ws   = (float*)d_ws;
    float* nP   = ws + 0 * NROWS;
    float* nQ   = ws + 1 * NROWS;
    float* nuQ  = ws + 2 * NROWS;
    float* rhoQ = ws + 3 * NROWS;
    float* nuP  = ws + 4 * NROWS;
    float* rhoP = ws + 5 * NROWS;
    float* kp1  = ws + 6 * NROWS;
    float* kp2  = ws + 7 * NROWS;

    norms_kernel<<<NROWS / 256, 256, 0, stream>>>(P, nP, NROWS);
    norms_kernel<<<NROWS / 256, 256, 0, stream>>>(Q, nQ, NROWS);

    // nu over Q (direction 1: estimate(P,Q)) and over P (direction 2)
    knn_nu_kernel<<<TILES, 32 * NWAVES, 0, stream>>>(Q, nQ, nuQ, rhoQ);
    knn_nu_kernel<<<TILES, 32 * NWAVES, 0, stream>>>(P, nP, nuP, rhoP);

    // k_p counts
    count_kernel<<<TILES, 32 * NWAVES, 0, stream>>>(P, nP, Q, nQ, rhoQ, kp1);
    count_kernel<<<TILES, 32 * NWAVES, 0, stream>>>(Q, nQ, P, nP, rhoP, kp2);

    finalize_kernel<<<1, 256, 0, stream>>>(kp1, nuQ, kp2, nuP, out);
}